// GNNModel_69260642615845
// MI455X (gfx1250) — compile-verified
//
#include <hip/hip_runtime.h>
#include <hip/hip_bf16.h>

// ---------------------------------------------------------------------------
// SR-GNN forward for MI455X (gfx1250).  All GEMMs use V_WMMA_F32_16X16X4_F32
// (keeps reference f32 semantics; workload is HBM-BW bound, ~1.5 GB traffic,
// so f32 WMMA throughput is far above what's needed).
// ---------------------------------------------------------------------------

typedef __attribute__((ext_vector_type(2))) float v2f;
typedef __attribute__((ext_vector_type(8))) float v8f;

static constexpr int H       = 256;
static constexpr int N_NODE  = 50000;
static constexpr int E_FULL  = 500000;
static constexpr int B_SESS  = 1024;
static constexpr int NNODES  = 12288;     // B * 12

// ---- workspace layout (float offsets) ----
static constexpr size_t OFF_BUFA  = 0;                       // m, then hidden_all (12.8M)
static constexpr size_t OFF_BUFB  = 12800000;                // agg, then embT   (12.8M)
static constexpr size_t OFF_GATE  = 25600000;                // 12288*256
static constexpr size_t OFF_WIHT  = 28745728;                // 256*768
static constexpr size_t OFF_WHHT  = 28942336;                // 256*768
static constexpr size_t OFF_W1T   = 29138944;                // 256*256
static constexpr size_t OFF_W2T   = 29204480;                // 256*256
static constexpr size_t OFF_W3T   = 29270016;                // 512*256
static constexpr size_t OFF_ALPHA = 29401088;                // 12288
static constexpr size_t OFF_LAST  = 29413376;                // 1024 (int)
static constexpr size_t OFF_VN    = 29414400;                // 1024*256
static constexpr size_t OFF_SG    = 29676544;                // 1024*256
static constexpr size_t OFF_SH    = 29938688;                // 1024*256
static constexpr size_t OFF_CAT   = 30200832;                // 1024*512
// total ~30.73M floats = ~123 MB

// ---- d_out layout ----
static constexpr size_t OUT_SCORES = 0;                          // 1024*50000
static constexpr size_t OUT_H2     = (size_t)B_SESS * N_NODE;    // 12288*256
static constexpr size_t OUT_EDGE   = OUT_H2 + (size_t)NNODES * H;// 2*12288

// fast transcendental helpers (VALU-cheap: v_exp + v_rcp, no IEEE div chain)
__device__ __forceinline__ float sigf(float x) {
  return __builtin_amdgcn_rcpf(1.0f + __expf(-x));
}
__device__ __forceinline__ float tanh_fast(float x) {
  float xc = fminf(fmaxf(x, -15.0f), 15.0f);
  float t = __expf(2.0f * xc);
  return (t - 1.0f) * __builtin_amdgcn_rcpf(t + 1.0f);
}

// WMMA f32 16x16x4 fragment helpers ----------------------------------------
// A tile lives in LDS as [16][stride] row-major.
__device__ __forceinline__ v2f fragA_lds(const float* As, int stride, int k,
                                         int laneLo, int hi) {
  v2f a;
  a.x = As[laneLo * stride + k + 2 * hi];
  a.y = As[laneLo * stride + k + 2 * hi + 1];
  return a;
}
// B row-major [K x ldb] in global memory; columns n0..n0+15.
__device__ __forceinline__ v2f fragB_g(const float* Bm, int ldb, int k, int n0,
                                       int laneLo, int hi) {
  v2f b;
  b.x = Bm[(size_t)(k + 2 * hi) * ldb + n0 + laneLo];
  b.y = Bm[(size_t)(k + 1 + 2 * hi) * ldb + n0 + laneLo];
  return b;
}
__device__ __forceinline__ v8f wmma4(v2f a, v2f b, v8f c) {
  return __builtin_amdgcn_wmma_f32_16x16x4_f32(false, a, false, b, (short)0, c,
                                               false, false);
}

// stage a 16 x K f32 tile (row-major, row stride = K) into LDS with row
// stride `stride` floats, using float4 (b128) transfers. 128 threads.
__device__ __forceinline__ void stage16(const float* __restrict__ src, int K,
                                        float* __restrict__ dstLds, int stride,
                                        int tid) {
  const int rowLen4 = K >> 2;
  const float4* s4 = (const float4*)src;
  for (int i4 = tid; i4 < 16 * rowLen4; i4 += 128) {
    int r = i4 / rowLen4, c4 = i4 - r * rowLen4;
    float4 v = s4[(size_t)r * rowLen4 + c4];
    *(float4*)&dstLds[r * stride + 4 * c4] = v;
  }
}

// ---------------------------------------------------------------------------
// Generic tiled transpose: dst[C x R] = src[R x C]^T
// ---------------------------------------------------------------------------
__global__ void transpose_k(const float* __restrict__ src, float* __restrict__ dst,
                            int R, int C) {
  __shared__ float t[32][33];
  int c0 = blockIdx.x * 32, r0 = blockIdx.y * 32;
  for (int dy = threadIdx.y; dy < 32; dy += 8) {
    int r = r0 + dy, c = c0 + threadIdx.x;
    if (r < R && c < C) t[dy][threadIdx.x] = src[(size_t)r * C + c];
  }
  __syncthreads();
  for (int dy = threadIdx.y; dy < 32; dy += 8) {
    int c = c0 + dy, r = r0 + threadIdx.x;
    if (r < R && c < C) dst[(size_t)c * R + r] = t[threadIdx.x][dy];
  }
}

// ---------------------------------------------------------------------------
// Generic WMMA GEMM: Out[M x N] = A[M x K] @ B[K x N] (+ bias[N]).
// block = 128 threads (4 waves); blockIdx.x = M/16 tile; each wave owns TPW
// consecutive 16-col tiles selected by blockIdx.y. A tile staged in LDS.
// ---------------------------------------------------------------------------
template <int TPW>
__global__ void gemm16_k(const float* __restrict__ A, const float* __restrict__ Bm,
                         const float* __restrict__ bias, float* __restrict__ Out,
                         int K, int Ntiles, int ldb, int ldo) {
  extern __shared__ float As[];               // 16 x (K+4)
  const int tid = threadIdx.x;
  const int wave = tid >> 5, lane = tid & 31;
  const int laneLo = lane & 15, hi = lane >> 4;
  const int stride = K + 4;
  const int m0 = blockIdx.x * 16;

  stage16(A + (size_t)m0 * K, K, As, stride, tid);
  __syncthreads();

  const int tile0 = (blockIdx.y * 4 + wave) * TPW;
  v8f z = {0.f, 0.f, 0.f, 0.f, 0.f, 0.f, 0.f, 0.f};
  v8f acc[TPW];
#pragma unroll
  for (int j = 0; j < TPW; ++j) acc[j] = z;

  for (int k = 0; k < K; k += 4) {
    v2f a = fragA_lds(As, stride, k, laneLo, hi);
#pragma unroll
    for (int j = 0; j < TPW; ++j) {
      int t = tile0 + j;
      if (t < Ntiles) {
        v2f b = fragB_g(Bm, ldb, k, t * 16, laneLo, hi);
        acc[j] = wmma4(a, b, acc[j]);
      }
    }
  }
#pragma unroll
  for (int j = 0; j < TPW; ++j) {
    int t = tile0 + j;
    if (t >= Ntiles) continue;
    int col = t * 16 + laneLo;
    float bb = bias ? bias[col] : 0.0f;
#pragma unroll
    for (int r = 0; r < 8; ++r) {
      int M = r + 8 * hi;
      Out[(size_t)(m0 + M) * ldo + col] = acc[j][r] + bb;
    }
  }
}

// ---------------------------------------------------------------------------
// Edge scatter-add: agg[dst] += m[src]   (one edge row per block, 256 lanes)
// ---------------------------------------------------------------------------
__global__ void segsum_k(const float* __restrict__ m, const int* __restrict__ fg,
                         float* __restrict__ agg) {
  int e = blockIdx.x, c = threadIdx.x;
  int s = fg[e];              // full_graph[0][e]
  int d = fg[E_FULL + e];     // full_graph[1][e]
  unsafeAtomicAdd(&agg[(size_t)d * H + c], m[(size_t)s * H + c]);
}

// ---------------------------------------------------------------------------
// Fused GRU cell: hidden = GRU(agg, emb).  Per block: one 16-node tile.
// gi = agg@WihT, gh = emb@WhhT computed per 16-col slice via WMMA (3 slices
// each for r/z/n), then elementwise GRU.  Avoids materializing 2x[50000x768].
// ---------------------------------------------------------------------------
__global__ void gru_k(const float* __restrict__ agg, const float* __restrict__ emb,
                      const float* __restrict__ WihT, const float* __restrict__ WhhT,
                      const float* __restrict__ bih, const float* __restrict__ bhh,
                      float* __restrict__ hidden) {
  __shared__ float Ag[16 * 260];
  __shared__ float Em[16 * 260];
  const int tid = threadIdx.x;
  const int wave = tid >> 5, lane = tid & 31;
  const int laneLo = lane & 15, hi = lane >> 4;
  const int m0 = blockIdx.x * 16;

  stage16(agg + (size_t)m0 * H, H, Ag, 260, tid);
  stage16(emb + (size_t)m0 * H, H, Em, 260, tid);
  __syncthreads();

  v8f z8 = {0.f, 0.f, 0.f, 0.f, 0.f, 0.f, 0.f, 0.f};
  for (int ci = wave; ci < 16; ci += 4) {     // 16-col slice of H
    v8f gi0 = z8, gi1 = z8, gi2 = z8, gh0 = z8, gh1 = z8, gh2 = z8;
    int n0 = ci * 16;
    for (int k = 0; k < H; k += 4) {
      v2f aA = fragA_lds(Ag, 260, k, laneLo, hi);
      v2f aE = fragA_lds(Em, 260, k, laneLo, hi);
      gi0 = wmma4(aA, fragB_g(WihT, 768, k, n0, laneLo, hi), gi0);
      gi1 = wmma4(aA, fragB_g(WihT, 768, k, n0 + 256, laneLo, hi), gi1);
      gi2 = wmma4(aA, fragB_g(WihT, 768, k, n0 + 512, laneLo, hi), gi2);
      gh0 = wmma4(aE, fragB_g(WhhT, 768, k, n0, laneLo, hi), gh0);
      gh1 = wmma4(aE, fragB_g(WhhT, 768, k, n0 + 256, laneLo, hi), gh1);
      gh2 = wmma4(aE, fragB_g(WhhT, 768, k, n0 + 512, laneLo, hi), gh2);
    }
    int col = n0 + laneLo;
    float bir = bih[col], biz = bih[col + 256], bin = bih[col + 512];
    float bhr = bhh[col], bhz = bhh[col + 256], bhn = bhh[col + 512];
#pragma unroll
    for (int r = 0; r < 8; ++r) {
      int M = r + 8 * hi;
      float rg = sigf(gi0[r] + bir + gh0[r] + bhr);
      float zg = sigf(gi1[r] + biz + gh1[r] + bhz);
      float ng = tanh_fast(gi2[r] + bin + rg * (gh2[r] + bhn));
      float e = Em[M * 260 + col];
      hidden[(size_t)(m0 + M) * H + col] = (1.0f - zg) * ng + zg * e;
    }
  }
}

// h2 = relu(hidden[x-1])  (one session-node row per block)
__global__ void gather_relu_k(const float* __restrict__ hidden,
                              const int* __restrict__ x, float* __restrict__ h2) {
  int i = blockIdx.x, c = threadIdx.x;
  int row = x[i] - 1;
  h2[(size_t)i * H + c] = fmaxf(hidden[(size_t)row * H + c], 0.0f);
}

__global__ void last_idx_k(const int* __restrict__ batch, int* __restrict__ last) {
  int i = blockIdx.x * blockDim.x + threadIdx.x;
  if (i < NNODES) {
    if (i == NNODES - 1 || batch[i + 1] != batch[i]) last[batch[i]] = i;
  }
}

__global__ void vn_k(const float* __restrict__ h2, const int* __restrict__ last,
                     float* __restrict__ vn) {
  int b = blockIdx.x, c = threadIdx.x;
  vn[(size_t)b * H + c] = h2[(size_t)last[b] * H + c];
}

// gate = sigmoid(vn[batch]@W1T + b1 + h2@W2T + b2), fused two-GEMM WMMA
__global__ void gate_k(const float* __restrict__ vn, const int* __restrict__ batch,
                       const float* __restrict__ h2, const float* __restrict__ W1T,
                       const float* __restrict__ W2T, const float* __restrict__ b1,
                       const float* __restrict__ b2, float* __restrict__ gate) {
  __shared__ float Vt[16 * 260];
  __shared__ float Ht[16 * 260];
  const int tid = threadIdx.x;
  const int wave = tid >> 5, lane = tid & 31;
  const int laneLo = lane & 15, hi = lane >> 4;
  const int m0 = blockIdx.x * 16;

  // rows of the vn-replicated tile come from v_n[batch[row]] (scattered); the
  // h2 tile is contiguous -> vectorized staging.
  for (int idx = tid; idx < 16 * H; idx += 128) {
    int r = idx >> 8, c = idx & 255;
    Vt[r * 260 + c] = vn[(size_t)batch[m0 + r] * H + c];
  }
  stage16(h2 + (size_t)m0 * H, H, Ht, 260, tid);
  __syncthreads();

  v8f z8 = {0.f, 0.f, 0.f, 0.f, 0.f, 0.f, 0.f, 0.f};
  for (int ct = wave; ct < 16; ct += 4) {
    v8f acc = z8;
    int n0 = ct * 16;
    for (int k = 0; k < H; k += 4) {
      v2f aV = fragA_lds(Vt, 260, k, laneLo, hi);
      v2f aH = fragA_lds(Ht, 260, k, laneLo, hi);
      acc = wmma4(aV, fragB_g(W1T, H, k, n0, laneLo, hi), acc);
      acc = wmma4(aH, fragB_g(W2T, H, k, n0, laneLo, hi), acc);
    }
    int col = n0 + laneLo;
    float bb = b1[col] + b2[col];
#pragma unroll
    for (int r = 0; r < 8; ++r) {
      int M = r + 8 * hi;
      gate[(size_t)(m0 + M) * H + col] = sigf(acc[r] + bb);
    }
  }
}

// alpha[i] = dot(gate[i], q_w) + q_b  (block-level tree reduction)
__global__ void alpha_k(const float* __restrict__ gate, const float* __restrict__ qw,
                        const float* __restrict__ qb, float* __restrict__ alpha) {
  __shared__ float red[256];
  int i = blockIdx.x, t = threadIdx.x;
  red[t] = gate[(size_t)i * H + t] * qw[t];
  __syncthreads();
  for (int s = 128; s > 0; s >>= 1) {
    if (t < s) red[t] += red[t + s];
    __syncthreads();
  }
  if (t == 0) alpha[i] = red[0] + qb[0];
}

// s_g[b] = sum_{i in segment b} alpha[i]*h2[i]   (segments are contiguous)
__global__ void sg_k(const float* __restrict__ alpha, const float* __restrict__ h2,
                     const int* __restrict__ last, float* __restrict__ sg) {
  int b = blockIdx.x, c = threadIdx.x;
  int st = (b == 0) ? 0 : last[b - 1] + 1;
  int en = last[b];
  float acc = 0.0f;
  for (int i = st; i <= en; ++i) acc += alpha[i] * h2[(size_t)i * H + c];
  sg[(size_t)b * H + c] = acc;
}

// cat = [v_n | s_g]
__global__ void cat_k(const float* __restrict__ vn, const float* __restrict__ sg,
                      float* __restrict__ cat) {
  int b = blockIdx.x, c = threadIdx.x;
  cat[(size_t)b * 2 * H + c] = vn[(size_t)b * H + c];
  cat[(size_t)b * 2 * H + H + c] = sg[(size_t)b * H + c];
}

// edge_index int -> float passthrough
__global__ void edge_k(const int* __restrict__ ei, float* __restrict__ out) {
  int i = blockIdx.x * blockDim.x + threadIdx.x;
  if (i < 2 * NNODES) out[i] = (float)ei[i];
}

// ---------------------------------------------------------------------------
extern "C" void kernel_launch(void* const* d_in, const int* in_sizes, int n_in,
                              void* d_out, int out_size, void* d_ws, size_t ws_size,
                              hipStream_t stream) {
  const int* x      = (const int*)d_in[0];
  const int* batch  = (const int*)d_in[1];
  const int* edge   = (const int*)d_in[2];
  const int* fg     = (const int*)d_in[3];
  const float* emb  = (const float*)d_in[4];
  const float* gW   = (const float*)d_in[5];
  const float* Wih  = (const float*)d_in[6];
  const float* Whh  = (const float*)d_in[7];
  const float* bih  = (const float*)d_in[8];
  const float* bhh  = (const float*)d_in[9];
  const float* W1w  = (const float*)d_in[10];
  const float* W1b  = (const float*)d_in[11];
  const float* W2w  = (const float*)d_in[12];
  const float* W2b  = (const float*)d_in[13];
  const float* qw   = (const float*)d_in[14];
  const float* qb   = (const float*)d_in[15];
  const float* W3w  = (const float*)d_in[16];
  const float* W3b  = (const float*)d_in[17];

  float* ws    = (float*)d_ws;
  float* bufA  = ws + OFF_BUFA;   // m, then hidden_all
  float* bufB  = ws + OFF_BUFB;   // agg, then embT
  float* gate  = ws + OFF_GATE;
  float* WihT  = ws + OFF_WIHT;
  float* WhhT  = ws + OFF_WHHT;
  float* W1T   = ws + OFF_W1T;
  float* W2T   = ws + OFF_W2T;
  float* W3T   = ws + OFF_W3T;
  float* alpha = ws + OFF_ALPHA;
  int*   last  = (int*)(ws + OFF_LAST);
  float* vn    = ws + OFF_VN;
  float* sg    = ws + OFF_SG;
  float* sh    = ws + OFF_SH;
  float* cat   = ws + OFF_CAT;

  float* scores = (float*)d_out + OUT_SCORES;
  float* h2     = (float*)d_out + OUT_H2;
  float* eout   = (float*)d_out + OUT_EDGE;

  // 1) weight transposes (B matrices must be row-major [K x N])
  transpose_k<<<dim3(8, 24), dim3(32, 8), 0, stream>>>(Wih, WihT, 768, 256);
  transpose_k<<<dim3(8, 24), dim3(32, 8), 0, stream>>>(Whh, WhhT, 768, 256);
  transpose_k<<<dim3(8, 8),  dim3(32, 8), 0, stream>>>(W1w, W1T, 256, 256);
  transpose_k<<<dim3(8, 8),  dim3(32, 8), 0, stream>>>(W2w, W2T, 256, 256);
  transpose_k<<<dim3(16, 8), dim3(32, 8), 0, stream>>>(W3w, W3T, 256, 512);

  // 2) m = emb @ ggnn_W   [50000 x 256]
  {
    size_t sh_bytes = 16 * (H + 4) * sizeof(float);
    gemm16_k<4><<<dim3(N_NODE / 16, 1), 128, sh_bytes, stream>>>(
        emb, gW, nullptr, bufA, H, 16, H, H);
  }

  // 3) agg = segment_sum(m[src], dst)
  hipMemsetAsync(bufB, 0, (size_t)N_NODE * H * sizeof(float), stream);
  segsum_k<<<E_FULL, H, 0, stream>>>(bufA, fg, bufB);

  // 4) hidden_all = GRU(agg, emb)   (overwrites m in bufA)
  gru_k<<<N_NODE / 16, 128, 0, stream>>>(bufB, emb, WihT, WhhT, bih, bhh, bufA);

  // 5) embT for the scores GEMM (agg in bufB is dead now)
  transpose_k<<<dim3(8, 1563), dim3(32, 8), 0, stream>>>(emb, bufB, N_NODE, H);

  // 6) h2 = relu(hidden_all[x-1]) -> straight into d_out
  gather_relu_k<<<NNODES, H, 0, stream>>>(bufA, x, h2);

  // 7) attention readout
  last_idx_k<<<(NNODES + 255) / 256, 256, 0, stream>>>(batch, last);
  vn_k<<<B_SESS, H, 0, stream>>>(h2, last, vn);
  gate_k<<<NNODES / 16, 128, 0, stream>>>(vn, batch, h2, W1T, W2T, W1b, W2b, gate);
  alpha_k<<<NNODES, H, 0, stream>>>(gate, qw, qb, alpha);
  sg_k<<<B_SESS, H, 0, stream>>>(alpha, h2, last, sg);
  cat_k<<<B_SESS, H, 0, stream>>>(vn, sg, cat);

  // 8) s_h = cat @ W3T + b3   [1024 x 256], K = 512
  {
    size_t sh_bytes = 16 * (2 * H + 4) * sizeof(float);
    gemm16_k<4><<<dim3(B_SESS / 16, 1), 128, sh_bytes, stream>>>(
        cat, W3T, W3b, sh, 2 * H, 16, H, H);
  }

  // 9) scores = s_h @ embT   [1024 x 50000]
  {
    size_t sh_bytes = 16 * (H + 4) * sizeof(float);
    gemm16_k<8><<<dim3(B_SESS / 16, 98), 128, sh_bytes, stream>>>(
        sh, bufB, nullptr, scores, H, N_NODE / 16, N_NODE, N_NODE);
  }

  // 10) edge_index passthrough (int -> float)
  edge_k<<<(2 * NNODES + 255) / 256, 256, 0, stream>>>(edge, eout);
}